// MultiHeadAttention_6545530159627
// MI455X (gfx1250) — compile-verified
//
#include <hip/hip_runtime.h>

// Problem constants (from reference)
#define B_   4
#define C_   512
#define T_   1024
#define H_   8
#define K_   64
#define WIN_ 4

typedef __attribute__((ext_vector_type(2))) float v2f;
typedef __attribute__((ext_vector_type(8))) float v8f;

// CDNA5 native fp32 matrix op: D(16x16,f32) = A(16x4,f32) * B(4x16,f32) + C
// A layout: lane L(0-15): row L, K={0,1}; lane L+16: row L, K={2,3}
// B layout: VGPR0: K=0 (lanes 0-15) / K=2 (lanes 16-31); VGPR1: K=1 / K=3
// C/D layout: VGPR j, lanes 0-15: [M=j, N=lane]; lanes 16-31: [M=j+8, N=lane-16]
__device__ __forceinline__ v8f wmma4(v2f a, v2f b, v8f c) {
  return __builtin_amdgcn_wmma_f32_16x16x4_f32(
      /*neg_a=*/false, a, /*neg_b=*/false, b,
      /*c_mod=*/(short)0, c, /*reuse_a=*/false, /*reuse_b=*/false);
}

// ---------------------------------------------------------------------------
// Kernel 1: q/k/v projections.  Out[b,h,t,k] = (W @ In[b,:,t] + bias) * scale
// grid = (512 tiles, 3 matrices, B). One wave per 16(o) x 64(t) tile.
// ---------------------------------------------------------------------------
__global__ __launch_bounds__(32)
void qkv_proj_kernel(const float* __restrict__ x, const float* __restrict__ cin,
                     const float* __restrict__ Wq, const float* __restrict__ bq,
                     const float* __restrict__ Wk, const float* __restrict__ bk,
                     const float* __restrict__ Wv, const float* __restrict__ bv,
                     float* __restrict__ qws, float* __restrict__ kws,
                     float* __restrict__ vws)
{
  const int lane = threadIdx.x;
  const int lo = lane & 15, hi = lane >> 4;
  const int tileId = blockIdx.x;
  const int o0 = (tileId >> 4) * 16;   // 32 output-channel tiles
  const int t0 = (tileId & 15) * 64;   // 16 time strips of 64
  const int which = blockIdx.y;        // 0=q(x), 1=k(c), 2=v(c)
  const int b = blockIdx.z;

  const float* In   = (which == 0) ? x  : cin;
  const float* W    = (which == 0) ? Wq : (which == 1) ? Wk : Wv;
  const float* bias = (which == 0) ? bq : (which == 1) ? bk : bv;
  float* Out        = (which == 0) ? qws : (which == 1) ? kws : vws;
  const float scale = (which == 0) ? 0.125f : 1.0f;   // 1/sqrt(K)

  v8f acc[4] = {};
  const float* inB = In + (size_t)b * C_ * T_;
  #pragma unroll 4
  for (int kk = 0; kk < C_ / 4; ++kk) {
    const int crow = kk * 4 + 2 * hi;
    v2f a = *(const v2f*)(W + (size_t)(o0 + lo) * C_ + crow);   // W[o,c],W[o,c+1]
    const float* xp = inB + (size_t)crow * T_ + t0 + lo;        // In[b,crow,t0+lo]
    #pragma unroll
    for (int i = 0; i < 4; ++i) {
      v2f bm;
      bm.x = xp[i * 16];
      bm.y = xp[T_ + i * 16];
      acc[i] = wmma4(a, bm, acc[i]);
    }
  }

  float* ob = Out + (size_t)b * H_ * T_ * K_;
  #pragma unroll
  for (int i = 0; i < 4; ++i) {
    #pragma unroll
    for (int j = 0; j < 8; ++j) {
      const int o = o0 + j + 8 * hi;
      const int t = t0 + i * 16 + lo;
      const float val = (acc[i][j] + bias[o]) * scale;
      ob[((size_t)(o >> 6) * T_ + t) * K_ + (o & 63)] = val;    // [h][t][k]
    }
  }
}

// ---------------------------------------------------------------------------
// Kernel 2: attention for one (b, h, 16-query tile). One wave per workgroup.
// Full 16 x 1024 score row kept in LDS (row stride 1028 for bank skew +
// 16B alignment); exact 2-pass softmax; banded rel-K/rel-V corrections.
// ---------------------------------------------------------------------------
__global__ __launch_bounds__(32)
void attn_kernel(const float* __restrict__ qws, const float* __restrict__ kws,
                 const float* __restrict__ vws, const int* __restrict__ mask,
                 const float* __restrict__ relk, const float* __restrict__ relv,
                 float* __restrict__ att)
{
  extern __shared__ float smem[];
  const int SROW = 1028;
  float* sS    = smem;                   // [16][1028] scores / probabilities
  float* sRelK = smem + 16 * SROW;       // [16][64] (rows 9..15 zero)
  float* sRelV = sRelK + 16 * 64;        // [16][64] (rows 9..15 zero)

  const int lane = threadIdx.x, lo = lane & 15, hi = lane >> 4;
  const int t0 = blockIdx.x * 16;
  const int h = blockIdx.y, b = blockIdx.z;
  const size_t bh = (size_t)(b * H_ + h) * T_ * K_;

  // Stage rel embeddings, zero-padded to 16 rows so they can be WMMA B tiles.
  for (int i = lane; i < 16 * 64; i += 32) {
    const int r = i >> 6, cc = i & 63;
    sRelK[i] = (r < 2 * WIN_ + 1) ? relk[r * K_ + cc] : 0.0f;
    sRelV[i] = (r < 2 * WIN_ + 1) ? relv[r * K_ + cc] : 0.0f;
  }
  __syncthreads();

  // Q tile (16 x 64) resident in registers in WMMA-A layout.
  v2f qreg[16];
  const float* Qp = qws + bh + (size_t)(t0 + lo) * K_ + 2 * hi;
  #pragma unroll
  for (int kk = 0; kk < 16; ++kk) qreg[kk] = *(const v2f*)(Qp + kk * 4);

  // Pass 1: S = Q K^T for all 64 key tiles -> LDS.
  for (int st = 0; st < T_ / 16; ++st) {
    const int s0 = st * 16;
    const float* Kp = kws + bh + (size_t)(s0 + lo) * K_ + 2 * hi;
    v8f acc = {};
    #pragma unroll
    for (int kk = 0; kk < 16; ++kk) {
      v2f bm = *(const v2f*)(Kp + kk * 4);          // K[s0+lo, kk4+2hi..+1]
      acc = wmma4(qreg[kk], bm, acc);
    }
    #pragma unroll
    for (int j = 0; j < 8; ++j)
      sS[(j + 8 * hi) * SROW + s0 + lo] = acc[j];
  }

  // Rel-K logits rl[m,d] = sum_k Q[m,k]*emb_rel_k[d,k]; scatter into band.
  v8f rl = {};
  #pragma unroll
  for (int kk = 0; kk < 16; ++kk) {
    v2f bm = *(const v2f*)(&sRelK[lo * 64 + kk * 4 + 2 * hi]);
    rl = wmma4(qreg[kk], bm, rl);
  }
  __syncthreads();
  if (lo < 2 * WIN_ + 1) {                          // d = lo in [0,8]
    #pragma unroll
    for (int j = 0; j < 8; ++j) {
      const int m = j + 8 * hi;
      const int s = t0 + m + lo - WIN_;             // s = t + d - WIN
      if (s >= 0 && s < T_) sS[m * SROW + s] += rl[j];
    }
  }
  __syncthreads();

  // Mask (reference: where(mask==0, -1e4, scores)).
  const int* Mp = mask + ((size_t)b * T_ + t0) * T_;
  for (int i = lane; i < 16 * T_; i += 32) {
    const int r = i >> 10, cc = i & (T_ - 1);
    if (Mp[(size_t)r * T_ + cc] == 0) sS[r * SROW + cc] = -1e4f;
  }
  __syncthreads();

  // Exact softmax over each full row (row fits in LDS -> no online rescale).
  for (int r = 0; r < 16; ++r) {
    float mx = -3.0e38f;
    for (int cc = lane; cc < T_; cc += 32) mx = fmaxf(mx, sS[r * SROW + cc]);
    #pragma unroll
    for (int off = 16; off > 0; off >>= 1) mx = fmaxf(mx, __shfl_xor(mx, off, 32));
    float sum = 0.0f;
    for (int cc = lane; cc < T_; cc += 32) {
      const float e = __expf(sS[r * SROW + cc] - mx);
      sS[r * SROW + cc] = e;
      sum += e;
    }
    #pragma unroll
    for (int off = 16; off > 0; off >>= 1) sum += __shfl_xor(sum, off, 32);
    const float inv = 1.0f / sum;
    for (int cc = lane; cc < T_; cc += 32) sS[r * SROW + cc] *= inv;
  }
  __syncthreads();

  // Pass 2: out = P V  (P tiles read from LDS in WMMA-A layout).
  v8f oacc[4] = {};
  const float* Vp = vws + bh;
  for (int st = 0; st < T_ / 16; ++st) {
    const int s0 = st * 16;
    #pragma unroll
    for (int ss = 0; ss < 4; ++ss) {
      const int sb = s0 + ss * 4 + 2 * hi;
      v2f a = *(const v2f*)(&sS[lo * SROW + sb]);   // P[lo, sb], P[lo, sb+1]
      const float* vr0 = Vp + (size_t)sb * K_ + lo;
      const float* vr1 = vr0 + K_;
      #pragma unroll
      for (int i = 0; i < 4; ++i) {
        v2f bm;
        bm.x = vr0[i * 16];                          // V[sb,   n0+lo]
        bm.y = vr1[i * 16];                          // V[sb+1, n0+lo]
        oacc[i] = wmma4(a, bm, oacc[i]);
      }
    }
  }

  // Rel-V band: out[t,k] += sum_d P[t, t+d-WIN] * emb_rel_v[d,k]
  #pragma unroll
  for (int n0i = 0; n0i < 4; ++n0i) {
    float rv[2 * WIN_ + 1];
    #pragma unroll
    for (int d = 0; d < 2 * WIN_ + 1; ++d) rv[d] = sRelV[d * 64 + n0i * 16 + lo];
    #pragma unroll
    for (int j = 0; j < 8; ++j) {
      const int m = j + 8 * hi;
      const int t = t0 + m;
      float add = 0.0f;
      #pragma unroll
      for (int d = 0; d < 2 * WIN_ + 1; ++d) {
        const int s = t + d - WIN_;
        if (s >= 0 && s < T_) add += sS[m * SROW + s] * rv[d];
      }
      oacc[n0i][j] += add;
    }
  }

  // Store attention output in [B,H,T,K] for the final projection.
  float* Op = att + bh;
  #pragma unroll
  for (int n0i = 0; n0i < 4; ++n0i)
    #pragma unroll
    for (int j = 0; j < 8; ++j)
      Op[(size_t)(t0 + j + 8 * hi) * K_ + n0i * 16 + lo] = oacc[n0i][j];
}

// ---------------------------------------------------------------------------
// Kernel 3: output projection. out[b,o,t] = Wo @ att + bo.
// att is [B,H,T,64] -> consecutive c within a head are contiguous float2.
// ---------------------------------------------------------------------------
__global__ __launch_bounds__(32)
void out_proj_kernel(const float* __restrict__ att, const float* __restrict__ Wo,
                     const float* __restrict__ bo, float* __restrict__ out)
{
  const int lane = threadIdx.x, lo = lane & 15, hi = lane >> 4;
  const int tileId = blockIdx.x;
  const int o0 = (tileId >> 4) * 16;
  const int t0 = (tileId & 15) * 64;
  const int b = blockIdx.y;

  v8f acc[4] = {};
  const float* ab = att + (size_t)b * H_ * T_ * K_;
  #pragma unroll 4
  for (int kk = 0; kk < C_ / 4; ++kk) {
    const int crow = kk * 4 + 2 * hi;
    v2f a = *(const v2f*)(Wo + (size_t)(o0 + lo) * C_ + crow);
    const float* ap = ab + (size_t)(crow >> 6) * T_ * K_ + (crow & 63);
    #pragma unroll
    for (int i = 0; i < 4; ++i) {
      v2f bm = *(const v2f*)(ap + (size_t)(t0 + i * 16 + lo) * K_);
      acc[i] = wmma4(a, bm, acc[i]);
    }
  }
  #pragma unroll
  for (int i = 0; i < 4; ++i)
    #pragma unroll
    for (int j = 0; j < 8; ++j) {
      const int o = o0 + j + 8 * hi;
      const int t = t0 + i * 16 + lo;
      out[((size_t)b * C_ + o) * T_ + t] = acc[i][j] + bo[o];
    }
}

// ---------------------------------------------------------------------------
extern "C" void kernel_launch(void* const* d_in, const int* in_sizes, int n_in,
                              void* d_out, int out_size, void* d_ws, size_t ws_size,
                              hipStream_t stream) {
  (void)in_sizes; (void)n_in; (void)out_size; (void)ws_size;
  const float* x    = (const float*)d_in[0];
  const float* c    = (const float*)d_in[1];
  const int*   mask = (const int*)d_in[2];
  const float* Wq   = (const float*)d_in[3];
  const float* bq   = (const float*)d_in[4];
  const float* Wk   = (const float*)d_in[5];
  const float* bk   = (const float*)d_in[6];
  const float* Wv   = (const float*)d_in[7];
  const float* bv   = (const float*)d_in[8];
  const float* Wo   = (const float*)d_in[9];
  const float* bo   = (const float*)d_in[10];
  const float* relk = (const float*)d_in[11];
  const float* relv = (const float*)d_in[12];
  float* out = (float*)d_out;

  const size_t BHTK = (size_t)B_ * H_ * T_ * K_;   // 2M floats each
  float* qws = (float*)d_ws;
  float* kws = qws + BHTK;
  float* vws = kws + BHTK;
  float* att = vws + BHTK;                         // total 32 MB workspace

  dim3 g1(512, 3, B_);
  qkv_proj_kernel<<<g1, 32, 0, stream>>>(x, c, Wq, bq, Wk, bk, Wv, bv,
                                         qws, kws, vws);

  const size_t lds = (size_t)(16 * 1028 + 2 * 16 * 64) * sizeof(float); // ~74 KB
  dim3 g2(T_ / 16, H_, B_);
  attn_kernel<<<g2, 32, lds, stream>>>(qws, kws, vws, mask, relk, relv, att);

  dim3 g3(512, B_);
  out_proj_kernel<<<g3, 32, 0, stream>>>(att, Wo, bo, out);
}